// HybridPooler_6030134084096
// MI455X (gfx1250) — compile-verified
//
#include <hip/hip_runtime.h>
#include <math.h>

#define NB   16
#define NS   4096
#define NSP1 4097
#define NH   768
#define NH3  2304
#define NCH  16      // s-chunks in pass 2
#define SCH  256     // NS / NCH
#define HTW  256     // h-tile width
#define LNEPS 1e-5f
#define FINF __builtin_inff()

typedef __attribute__((ext_vector_type(2))) float v2f;
typedef __attribute__((ext_vector_type(8))) float v8f;

#if __has_builtin(__builtin_amdgcn_wmma_f32_16x16x4_f32)
#define HAVE_WMMA_F32X4 1
#endif

// -------- workspace layout (float offsets) --------
#define OFF_QG      0u        // float2 qg[768] : (q0*g, q1*g)
#define OFF_QC      1536u     // 4 scalars: qgs0, qgs1, qb0, qb1
#define OFF_STATS   2048u     // float2 per row (mu, rsig), 65536 rows
#define OFF_LOGITS  133120u   // [(b*2+q)*S + s]; overwritten with c = attn*rsig
#define OFF_DSCALE  264192u   // 32 scalars d[b][q]
#define OFF_PSUM    264704u   // partials: [B][NCH][H] each
#define OFF_PMAX    461312u
#define OFF_PMIN    657920u
#define OFF_PA0     854528u
#define OFF_PA1     1051136u
#define OFF_PTRAD   1247744u  // [16][2304] mean|max|min
#define OFF_PLEARN  1284608u  // [16][2304] pmp0|pmp1|clf
// total 1321472 floats ~= 5.3 MB

__device__ __forceinline__ float block_sum(float v, float* sh) {
  int t = threadIdx.x;
  sh[t] = v; __syncthreads();
  for (int o = 128; o > 0; o >>= 1) { if (t < o) sh[t] += sh[t + o]; __syncthreads(); }
  float r = sh[0]; __syncthreads();
  return r;
}

// ---- kernel 0: precompute qg[h] = q*g and scalars Σqg, Σq*b ----
__global__ void __launch_bounds__(256) prep_kernel(
    const float* __restrict__ q, const float* __restrict__ g,
    const float* __restrict__ bln, float* __restrict__ ws) {
  __shared__ float sh[256];
  int t = threadIdx.x;
  float s0 = 0.f, s1 = 0.f, e0 = 0.f, e1 = 0.f;
  for (int h = t; h < NH; h += 256) {
    float gh = g[h], q0 = q[h], q1 = q[NH + h], bh = bln[h];
    float2 v; v.x = q0 * gh; v.y = q1 * gh;
    ((float2*)(ws + OFF_QG))[h] = v;
    s0 += v.x; s1 += v.y; e0 += q0 * bh; e1 += q1 * bh;
  }
  s0 = block_sum(s0, sh); s1 = block_sum(s1, sh);
  e0 = block_sum(e0, sh); e1 = block_sum(e1, sh);
  if (t == 0) {
    ws[OFF_QC + 0] = s0; ws[OFF_QC + 1] = s1;
    ws[OFF_QC + 2] = e0; ws[OFF_QC + 3] = e1;
  }
}

// ---- kernel 1 (pass 1): one wave per token row -> (mu,rsig) + logits ----
__global__ void __launch_bounds__(256) row_stats_kernel(
    const float* __restrict__ tokens, const int* __restrict__ lengths,
    float* __restrict__ ws) {
  __shared__ float2 qg[NH];
  int t = threadIdx.x;
  for (int h = t; h < NH; h += 256) qg[h] = ((const float2*)(ws + OFF_QG))[h];
  __syncthreads();
  int wave = t >> 5, lane = t & 31;
  int r = blockIdx.x * 8 + wave;          // row in [0, B*S)
  int b = r >> 12, s = r & 4095;
  if (s >= lengths[b]) return;            // skip invalid rows (bandwidth saver)
  const float qgs0 = ws[OFF_QC + 0], qgs1 = ws[OFF_QC + 1];
  const float qb0  = ws[OFF_QC + 2], qb1  = ws[OFF_QC + 3];
  const float* xr = tokens + ((size_t)(b * NSP1 + 1 + s)) * NH;
  float s1 = 0.f, s2 = 0.f, d0 = 0.f, d1 = 0.f;
#pragma unroll
  for (int k = 0; k < 24; ++k) {
    int h = k * 32 + lane;
    float v = xr[h];
    float2 w = qg[h];
    s1 += v; s2 += v * v; d0 += v * w.x; d1 += v * w.y;
  }
#pragma unroll
  for (int o = 16; o > 0; o >>= 1) {
    s1 += __shfl_xor(s1, o, 32); s2 += __shfl_xor(s2, o, 32);
    d0 += __shfl_xor(d0, o, 32); d1 += __shfl_xor(d1, o, 32);
  }
  if (lane == 0) {
    float mu = s1 * (1.0f / NH);
    float var = s2 * (1.0f / NH) - mu * mu;
    float rs = rsqrtf(var + LNEPS);
    ((float2*)(ws + OFF_STATS))[r] = make_float2(mu, rs);
    const float isq = 0.03608439182435161f;   // 768^-0.5
    ws[OFF_LOGITS + (size_t)(b * 2 + 0) * NS + s] = ((d0 - mu * qgs0) * rs + qb0) * isq;
    ws[OFF_LOGITS + (size_t)(b * 2 + 1) * NS + s] = ((d1 - mu * qgs1) * rs + qb1) * isq;
  }
}

// ---- kernel 2: masked softmax; emit c[s]=attn*rsig in place, d=Σattn*mu*rsig ----
__global__ void __launch_bounds__(256) softmax_kernel(
    const int* __restrict__ lengths, float* __restrict__ ws) {
  __shared__ float sh[256];
  int t = threadIdx.x, bq = blockIdx.x, b = bq >> 1;
  int len = lengths[b];
  float* row = ws + OFF_LOGITS + (size_t)bq * NS;
  const float2* st = ((const float2*)(ws + OFF_STATS)) + (size_t)b * NS;
  float m = -FINF;
  for (int s = t; s < len; s += 256) m = fmaxf(m, row[s]);
  sh[t] = m; __syncthreads();
  for (int o = 128; o > 0; o >>= 1) { if (t < o) sh[t] = fmaxf(sh[t], sh[t + o]); __syncthreads(); }
  m = sh[0]; __syncthreads();
  float z = 0.f;
  for (int s = t; s < len; s += 256) z += expf(row[s] - m);
  z = block_sum(z, sh);
  float invz = 1.0f / z, d = 0.f;
  for (int s = t; s < NS; s += 256) {
    float c = 0.0f;
    if (s < len) {
      float a = expf(row[s] - m) * invz;
      float2 ms = st[s];
      c = a * ms.y;
      d += a * ms.x * ms.y;
    }
    row[s] = c;
  }
  d = block_sum(d, sh);
  if (t == 0) ws[OFF_DSCALE + bq] = d;
}

// ---- kernel 3 (pass 2): fused pooling + weighted column sums, chunked ----
__global__ void __launch_bounds__(256) pass2_kernel(
    const float* __restrict__ tokens, const int* __restrict__ lengths,
    float* __restrict__ ws) {
  __shared__ float cc0[SCH], cc1[SCH];
  int t = threadIdx.x, bx = blockIdx.x;
  int b = bx / 48, rem = bx % 48;
  int ht = rem >> 4, ch = rem & 15;
  int len = lengths[b];
  int s0 = ch * SCH;
  int h = ht * HTW + t;
  cc0[t] = ws[OFF_LOGITS + (size_t)(b * 2 + 0) * NS + s0 + t];
  cc1[t] = ws[OFF_LOGITS + (size_t)(b * 2 + 1) * NS + s0 + t];
  __syncthreads();
  float psum = 0.f, pmax = -FINF, pmin = FINF, a0 = 0.f, a1 = 0.f;
  int nv = len - s0; if (nv > SCH) nv = SCH;
  const float* xp = tokens + ((size_t)(b * NSP1 + 1 + s0)) * NH + h;
  for (int i = 0; i < nv; ++i) {
    if (i + 8 < nv) __builtin_prefetch(xp + (size_t)(i + 8) * NH, 0, 3);
    float v = xp[(size_t)i * NH];
    psum += v;
    pmax = fmaxf(pmax, v);
    pmin = fminf(pmin, v);
    a0 += cc0[i] * v;
    a1 += cc1[i] * v;
  }
  size_t off = ((size_t)(b * NCH + ch)) * NH + h;
  ws[OFF_PSUM + off] = psum;
  ws[OFF_PMAX + off] = pmax;
  ws[OFF_PMIN + off] = pmin;
  ws[OFF_PA0 + off] = a0;
  ws[OFF_PA1 + off] = a1;
}

// ---- kernel 3b: combine chunks, assemble pooled_trad / pooled_learn ----
__global__ void __launch_bounds__(256) finalize_pool_kernel(
    const float* __restrict__ tokens, const int* __restrict__ lengths,
    const float* __restrict__ g, const float* __restrict__ bln,
    float* __restrict__ ws) {
  int idx = blockIdx.x * 256 + threadIdx.x;   // [0, 16*768)
  int b = idx / NH, h = idx % NH;
  float sum = 0.f, mx = -FINF, mn = FINF, A0 = 0.f, A1 = 0.f;
  for (int ch = 0; ch < NCH; ++ch) {
    size_t off = ((size_t)(b * NCH + ch)) * NH + h;
    sum += ws[OFF_PSUM + off];
    mx = fmaxf(mx, ws[OFF_PMAX + off]);
    mn = fminf(mn, ws[OFF_PMIN + off]);
    A0 += ws[OFF_PA0 + off];
    A1 += ws[OFF_PA1 + off];
  }
  float* pt = ws + OFF_PTRAD + (size_t)b * NH3;
  pt[h] = sum / (float)lengths[b];
  pt[NH + h] = mx;
  pt[2 * NH + h] = mn;
  float d0 = ws[OFF_DSCALE + b * 2 + 0], d1 = ws[OFF_DSCALE + b * 2 + 1];
  float gh = g[h], bh = bln[h];
  float* pl = ws + OFF_PLEARN + (size_t)b * NH3;
  pl[h] = gh * (A0 - d0) + bh;
  pl[NH + h] = gh * (A1 - d1) + bh;
  pl[2 * NH + h] = tokens[((size_t)b * NSP1) * NH + h];   // clf token
}

// ---- kernel 4: MLP via V_WMMA_F32_16X16X4_F32 (M=16 == batch) ----
__global__ void __launch_bounds__(256) mlp_kernel(
    const float* __restrict__ X,      // [16][2304]
    const float* __restrict__ Wa,     // [2304][768] row-major (K-major)
    const float* __restrict__ ba,
    const float* __restrict__ Wb,     // [768][768]
    const float* __restrict__ bb,
    float* __restrict__ out, int outOff) {
  __shared__ float hid[16][NH + 8];
  int t = threadIdx.x;
  int wave = t >> 5, lane = t & 31;
  int half = lane >> 4, tl = lane & 15;

  // GEMM1: (16 x 2304) @ (2304 x 768), 48 N-tiles, 6 per wave; + bias + exact GELU
  for (int j = 0; j < 6; ++j) {
    int n0 = (wave + 8 * j) * 16;
    v8f c = {};
    for (int k = 0; k < NH3; k += 4) {
      // A 16x4 f32: lanes 0-15 -> K=k,k+1 ; lanes 16-31 -> K=k+2,k+3 (M = tl)
      v2f a, bv;
      a.x = X[(size_t)tl * NH3 + k + 2 * half];
      a.y = X[(size_t)tl * NH3 + k + 2 * half + 1];
      // B 4x16 f32: VGPR0 rows K={0|2}, VGPR1 rows K={1|3}; N = n0+tl
      bv.x = Wa[(size_t)(k + 2 * half) * NH + n0 + tl];
      bv.y = Wa[(size_t)(k + 2 * half + 1) * NH + n0 + tl];
#ifdef HAVE_WMMA_F32X4
      c = __builtin_amdgcn_wmma_f32_16x16x4_f32(false, a, false, bv,
                                                (short)0, c, false, false);
#else
#pragma unroll
      for (int i = 0; i < 8; ++i) {
        c[i] += X[(size_t)(i + 8 * half) * NH3 + k] * Wa[(size_t)k * NH + n0 + tl];
        c[i] += X[(size_t)(i + 8 * half) * NH3 + k + 1] * Wa[(size_t)(k + 1) * NH + n0 + tl];
        c[i] += X[(size_t)(i + 8 * half) * NH3 + k + 2] * Wa[(size_t)(k + 2) * NH + n0 + tl];
        c[i] += X[(size_t)(i + 8 * half) * NH3 + k + 3] * Wa[(size_t)(k + 3) * NH + n0 + tl];
      }
#endif
    }
    float bias = ba[n0 + tl];
#pragma unroll
    for (int i = 0; i < 8; ++i) {
      float x = c[i] + bias;                       // D VGPR i: row i+8*half, col n0+tl
      hid[i + 8 * half][n0 + tl] = 0.5f * x * (1.0f + erff(x * 0.70710678118654752f));
    }
  }
  __syncthreads();

  // GEMM2: (16 x 768) @ (768 x 768) from LDS + bias -> out slice
  for (int j = 0; j < 6; ++j) {
    int n0 = (wave + 8 * j) * 16;
    v8f c = {};
    for (int k = 0; k < NH; k += 4) {
      v2f a, bv;
      a.x = hid[tl][k + 2 * half];
      a.y = hid[tl][k + 2 * half + 1];
      bv.x = Wb[(size_t)(k + 2 * half) * NH + n0 + tl];
      bv.y = Wb[(size_t)(k + 2 * half + 1) * NH + n0 + tl];
#ifdef HAVE_WMMA_F32X4
      c = __builtin_amdgcn_wmma_f32_16x16x4_f32(false, a, false, bv,
                                                (short)0, c, false, false);
#else
#pragma unroll
      for (int i = 0; i < 8; ++i) {
        c[i] += hid[i + 8 * half][k]     * Wb[(size_t)k * NH + n0 + tl];
        c[i] += hid[i + 8 * half][k + 1] * Wb[(size_t)(k + 1) * NH + n0 + tl];
        c[i] += hid[i + 8 * half][k + 2] * Wb[(size_t)(k + 2) * NH + n0 + tl];
        c[i] += hid[i + 8 * half][k + 3] * Wb[(size_t)(k + 3) * NH + n0 + tl];
      }
#endif
    }
    float bias = bb[n0 + tl];
#pragma unroll
    for (int i = 0; i < 8; ++i)
      out[(size_t)(i + 8 * half) * 1536 + outOff + n0 + tl] = c[i] + bias;
  }
}

extern "C" void kernel_launch(void* const* d_in, const int* in_sizes, int n_in,
                              void* d_out, int out_size, void* d_ws, size_t ws_size,
                              hipStream_t stream) {
  (void)in_sizes; (void)n_in; (void)out_size; (void)ws_size;
  const float* tokens = (const float*)d_in[0];
  const int*   lengths = (const int*)d_in[1];
  const float* q    = (const float*)d_in[2];
  const float* ln_g = (const float*)d_in[3];
  const float* ln_b = (const float*)d_in[4];
  const float* w1a  = (const float*)d_in[5];
  const float* b1a  = (const float*)d_in[6];
  const float* w1b  = (const float*)d_in[7];
  const float* b1b  = (const float*)d_in[8];
  const float* w2a  = (const float*)d_in[9];
  const float* b2a  = (const float*)d_in[10];
  const float* w2b  = (const float*)d_in[11];
  const float* b2b  = (const float*)d_in[12];
  float* out = (float*)d_out;
  float* ws  = (float*)d_ws;

  prep_kernel<<<1, 256, 0, stream>>>(q, ln_g, ln_b, ws);
  row_stats_kernel<<<(NB * NS) / 8, 256, 0, stream>>>(tokens, lengths, ws);
  softmax_kernel<<<NB * 2, 256, 0, stream>>>(lengths, ws);
  pass2_kernel<<<NB * 48, 256, 0, stream>>>(tokens, lengths, ws);
  finalize_pool_kernel<<<(NB * NH) / 256, 256, 0, stream>>>(tokens, lengths, ln_g, ln_b, ws);
  mlp_kernel<<<1, 256, 0, stream>>>(ws + OFF_PTRAD, w1a, b1a, w1b, b1b, out, 0);
  mlp_kernel<<<1, 256, 0, stream>>>(ws + OFF_PLEARN, w2a, b2a, w2b, b2b, out, NH);
}